// GATAUHead_45612552683745
// MI455X (gfx1250) — compile-verified
//
#include <hip/hip_runtime.h>
#include <math.h>

// ---------------- problem constants (match reference) ----------------
#define N_NODES   32
#define N_CH      256     // input channels = N_HEADS*HEAD_DIM
#define N_HEADS   4
#define HEAD_DIM  64
#define N_EDGE0   512
#define N_EDGE    544     // + self loops
#define N_AU      32
#define NEG_SLOPE 0.2f
#define PLANE     9216    // 96*96

// LDS strides chosen for bank-conflict-free WMMA fragment loads
#define XS_S 260          // 260 % 64 == 4  -> A-fragment b64 loads conflict-free
#define WP_S 272          // 272 % 64 == 16 -> B-fragment b32 loads conflict-free
#define KP   32           // K panel staged in LDS per iteration

typedef float v2f __attribute__((ext_vector_type(2)));
typedef float v8f __attribute__((ext_vector_type(8)));

// ===================================================================
// Kernel 1: spatial mean pool  [32,256,96,96] -> [32,256]
// One block per (n,c) plane; 256 threads x 9 float4 coalesced loads.
// ===================================================================
__global__ __launch_bounds__(256) void gat_pool_kernel(
    const float* __restrict__ rf, float* __restrict__ x)
{
    const int plane = blockIdx.x;                       // n*256 + c
    const float4* p = (const float4*)(rf + (size_t)plane * PLANE);
    float s = 0.f;
#pragma unroll
    for (int i = 0; i < 9; ++i) {
        float4 v = p[threadIdx.x + i * 256];
        s += v.x + v.y + v.z + v.w;
    }
    __shared__ float red[256];
    red[threadIdx.x] = s;
    __syncthreads();
    for (int off = 128; off > 0; off >>= 1) {
        if (threadIdx.x < off) red[threadIdx.x] += red[threadIdx.x + off];
        __syncthreads();
    }
    if (threadIdx.x == 0) x[plane] = red[0] * (1.0f / (float)PLANE);
}

// ===================================================================
// Kernel 2: fused GAT + classifier, one workgroup, 512 threads.
//   h = x@W via V_WMMA_F32_16X16X4_F32, attention softmax over edges,
//   scatter-sum via LDS float atomics, silu, h@cls_W.
// ===================================================================
__global__ __launch_bounds__(512) void gat_fused_kernel(
    const float* __restrict__ xg,        // [32,256] pooled (from d_ws)
    const int*   __restrict__ eidx,      // [2,512]
    const float* __restrict__ Wg,        // [256,256]
    const float* __restrict__ a_src,     // [4,64]
    const float* __restrict__ a_dst,     // [4,64]
    const float* __restrict__ gbias,     // [256]
    const float* __restrict__ clsW,      // [256,32]
    const float* __restrict__ clsb,      // [32]
    float*       __restrict__ out)       // [32,32]
{
    extern __shared__ float smem[];
    float* xs  = smem;                       // 32*XS_S
    float* hs  = xs  + N_NODES * XS_S;       // 32*XS_S
    float* ol  = hs  + N_NODES * XS_S;       // 32*XS_S (gat out / silu)
    float* wp  = ol  + N_NODES * XS_S;       // KP*WP_S (W panel)
    float* asl = wp  + KP * WP_S;            // 128 (alpha_src [n][h])
    float* adl = asl + 128;                  // 128
    float* ml  = adl + 128;                  // 128 (segment max)
    float* dl  = ml  + 128;                  // 128 (segment denom)
    float* ev  = dl  + 128;                  // 544*4 (edge score -> alpha)
    int*   es  = (int*)(ev + N_EDGE * 4);    // 544 (src)
    int*   ed  = es + N_EDGE;                // 544 (dst)

    const int tid = threadIdx.x;

    // ---- stage x into LDS, build edge lists (+ self loops) ----
    for (int idx = tid; idx < N_NODES * N_CH; idx += 512) {
        int n = idx >> 8, c = idx & 255;
        xs[n * XS_S + c] = xg[idx];
    }
    es[tid] = eidx[tid];                 // src row
    ed[tid] = eidx[N_EDGE0 + tid];       // dst row
    if (tid < N_NODES) { es[N_EDGE0 + tid] = tid; ed[N_EDGE0 + tid] = tid; }
    __syncthreads();

    // ---- GEMM h = x @ W  (32x256x256) with f32 WMMA 16x16x4 ----
    const int wave  = tid >> 5;          // 0..15 -> N tile (16 cols each)
    const int lane  = tid & 31;
    const int lm    = lane & 15;
    const int lh    = lane >> 4;         // 0: K+0..1, 1: K+2..3
    const int ncol  = wave * 16 + lm;

    v8f c0 = {0.f,0.f,0.f,0.f,0.f,0.f,0.f,0.f};
    v8f c1 = {0.f,0.f,0.f,0.f,0.f,0.f,0.f,0.f};

    for (int k0 = 0; k0 < N_CH; k0 += KP) {
        __syncthreads();                 // previous panel fully consumed
        // stage W[k0:k0+KP, 0:256] -> wp (float4 copies, 16B aligned)
        for (int idx = tid; idx < KP * 64; idx += 512) {
            int r  = idx >> 6;
            int c4 = idx & 63;
            float4 v = ((const float4*)(Wg + (size_t)(k0 + r) * N_CH))[c4];
            *(float4*)(wp + r * WP_S + c4 * 4) = v;
        }
        __syncthreads();
#pragma unroll
        for (int kk = 0; kk < KP; kk += 4) {
            const int ka = k0 + kk + lh * 2;     // A column (global k)
            const int kb = kk + lh * 2;          // B row (panel-local)
            v2f a0 = *(const v2f*)&xs[lm * XS_S + ka];
            v2f a1 = *(const v2f*)&xs[(16 + lm) * XS_S + ka];
            v2f b;
            b.x = wp[kb * WP_S + ncol];
            b.y = wp[(kb + 1) * WP_S + ncol];
            c0 = __builtin_amdgcn_wmma_f32_16x16x4_f32(false, a0, false, b,
                                                       (short)0, c0, false, false);
            c1 = __builtin_amdgcn_wmma_f32_16x16x4_f32(false, a1, false, b,
                                                       (short)0, c1, false, false);
        }
    }
    // D layout: VGPR r holds M = 8*lh + r
#pragma unroll
    for (int r = 0; r < 8; ++r) {
        hs[(lh * 8 + r) * XS_S + ncol]        = c0[r];
        hs[(16 + lh * 8 + r) * XS_S + ncol]   = c1[r];
    }
    __syncthreads();

    // ---- per-(node,head) attention coefficients ----
    if (tid < N_NODES * N_HEADS) {
        int n = tid >> 2, h = tid & 3;
        float ss = 0.f, sd = 0.f;
        for (int f = 0; f < HEAD_DIM; ++f) {
            float hv = hs[n * XS_S + h * HEAD_DIM + f];
            ss += hv * a_src[h * HEAD_DIM + f];
            sd += hv * a_dst[h * HEAD_DIM + f];
        }
        asl[tid] = ss;
        adl[tid] = sd;
    }
    __syncthreads();

    // ---- edge scores e = leaky_relu(as[src] + ad[dst]) ----
    for (int idx = tid; idx < N_EDGE * N_HEADS; idx += 512) {
        int e = idx >> 2, h = idx & 3;
        float v = asl[es[e] * 4 + h] + adl[ed[e] * 4 + h];
        ev[idx] = (v >= 0.f) ? v : NEG_SLOPE * v;
    }
    __syncthreads();

    // ---- segment softmax stats over dst ----
    if (tid < N_NODES * N_HEADS) {
        int n = tid >> 2, h = tid & 3;
        float mx = -INFINITY;
        for (int e = 0; e < N_EDGE; ++e)
            if (ed[e] == n) mx = fmaxf(mx, ev[e * 4 + h]);
        float sm = 0.f;
        for (int e = 0; e < N_EDGE; ++e)
            if (ed[e] == n) sm += __expf(ev[e * 4 + h] - mx);
        ml[tid] = mx;
        dl[tid] = sm;
    }
    __syncthreads();

    // ---- normalize to alpha (in place) ----
    for (int idx = tid; idx < N_EDGE * N_HEADS; idx += 512) {
        int e = idx >> 2, h = idx & 3;
        int nh = ed[e] * 4 + h;
        ev[idx] = __expf(ev[idx] - ml[nh]) / dl[nh];
    }
    // zero accumulator
    for (int idx = tid; idx < N_NODES * XS_S; idx += 512) ol[idx] = 0.f;
    __syncthreads();

    // ---- scatter: out[dst] += alpha * h[src]  (LDS ds_add_f32) ----
    for (int idx = tid; idx < N_EDGE * N_CH; idx += 512) {
        int e = idx >> 8, c = idx & 255;
        int h = c >> 6;
        float contrib = ev[e * 4 + h] * hs[es[e] * XS_S + c];
        atomicAdd(&ol[ed[e] * XS_S + c], contrib);
    }
    __syncthreads();

    // ---- bias + silu in place ----
    for (int idx = tid; idx < N_NODES * N_CH; idx += 512) {
        int n = idx >> 8, c = idx & 255;
        float v = ol[n * XS_S + c] + gbias[c];
        ol[n * XS_S + c] = v / (1.f + __expf(-v));
    }
    __syncthreads();

    // ---- classifier: logits = y @ cls_W + cls_b ----
    for (int idx = tid; idx < N_NODES * N_AU; idx += 512) {
        int n = idx >> 5, j = idx & 31;
        float acc = clsb[j];
        for (int c = 0; c < N_CH; ++c)
            acc += ol[n * XS_S + c] * clsW[c * N_AU + j];
        out[n * N_AU + j] = acc;
    }
}

// ===================================================================
extern "C" void kernel_launch(void* const* d_in, const int* in_sizes, int n_in,
                              void* d_out, int out_size, void* d_ws, size_t ws_size,
                              hipStream_t stream) {
    const float* roi   = (const float*)d_in[0];
    const int*   eidx  = (const int*)  d_in[1];
    const float* W     = (const float*)d_in[2];
    const float* a_src = (const float*)d_in[3];
    const float* a_dst = (const float*)d_in[4];
    const float* gbias = (const float*)d_in[5];
    const float* clsW  = (const float*)d_in[6];
    const float* clsb  = (const float*)d_in[7];
    float* out = (float*)d_out;
    float* x   = (float*)d_ws;           // [32,256] pooled features

    // Phase 1: HBM-bound mean pool (~302 MB -> ~13 us at 23.3 TB/s)
    gat_pool_kernel<<<N_NODES * N_CH, 256, 0, stream>>>(roi, x);

    // Phase 2: fused GAT + classifier, single WGP, dynamic LDS
    size_t lds = (size_t)(3 * N_NODES * XS_S + KP * WP_S + 4 * 128
                          + N_EDGE * 4) * sizeof(float)
               + (size_t)(2 * N_EDGE) * sizeof(int);
    gat_fused_kernel<<<1, 512, lds, stream>>>(x, eidx, W, a_src, a_dst,
                                              gbias, clsW, clsb, out);
}